// HopfRNNCellTheta_80221399154893
// MI455X (gfx1250) — compile-verified
//
#include <hip/hip_runtime.h>
#include <cstddef>

typedef __attribute__((ext_vector_type(2))) float v2f;
typedef __attribute__((ext_vector_type(8))) float v8f;

#define NN 4096

// ---------------------------------------------------------------------------
// Kernel 1: pack the complex mat-vec vectors (with signs folded in) into
// V[4][4096][2]:  per matrix plane m, column-0 (real-out) and column-1
// (imag-out) multiplier vectors.
//   A_re: ( z0_re,  z0_im)   A_im: (-z0_im,  z0_re)
//   B_re: ( x0_re,  x0_im)   B_im: (-x0_im,  x0_re)
// ---------------------------------------------------------------------------
__global__ void build_vectors(const float* __restrict__ x_re,
                              const float* __restrict__ x_im,
                              const float* __restrict__ z_re,
                              const float* __restrict__ z_im,
                              float* __restrict__ V) {
  int k = blockIdx.x * blockDim.x + threadIdx.x;
  if (k >= NN) return;
  float zr = z_re[(size_t)k * NN];
  float zi = z_im[(size_t)k * NN];
  float xr = x_re[(size_t)k * NN];
  float xi = x_im[(size_t)k * NN];
  V[(size_t)(0 * NN + k) * 2 + 0] = zr;
  V[(size_t)(0 * NN + k) * 2 + 1] = zi;
  V[(size_t)(1 * NN + k) * 2 + 0] = -zi;
  V[(size_t)(1 * NN + k) * 2 + 1] = zr;
  V[(size_t)(2 * NN + k) * 2 + 0] = xr;
  V[(size_t)(2 * NN + k) * 2 + 1] = xi;
  V[(size_t)(3 * NN + k) * 2 + 0] = -xi;
  V[(size_t)(3 * NN + k) * 2 + 1] = xr;
}

// ---------------------------------------------------------------------------
// Kernel 2: WMMA mat-vec.  Each wave owns a 16-row tile and a 256-wide K
// chunk; accumulator column 0 = real part, column 1 = imag part.
// A-matrix 16x4 f32 layout: lane L (0..15 and 16..31) -> M = L&15,
// VGPR0/1 -> K = (L>>4)*2 + {0,1}.  B-matrix 4x16: lane n carries column n,
// VGPR0/1 -> same K split.  Only columns 0,1 non-zero (cndmask, so EXEC
// stays all-ones around the WMMAs).
// partials layout: [row (4096)][kchunk (16)][re/im]
// ---------------------------------------------------------------------------
__global__ __launch_bounds__(256) void matvec_wmma(
    const float* __restrict__ A_re, const float* __restrict__ A_im,
    const float* __restrict__ B_re, const float* __restrict__ B_im,
    const float* __restrict__ V, float* __restrict__ partials) {
  const int lane = threadIdx.x & 31;
  const int wave = threadIdx.x >> 5;
  const int row0 = blockIdx.x * 16;
  const int chunk = blockIdx.y * 8 + wave;       // 0..15
  const int kbase = chunk * 256;

  const int nsel  = lane & 15;                   // row for A, column for B
  const int khalf = (lane >> 4) * 2;             // K offset for hi half-wave
  const int row   = row0 + nsel;
  const int bsel  = nsel & 1;                    // which packed component
  const bool bact = (nsel < 2);                  // only cols 0,1 are live

  const float* mats[4] = {A_re, A_im, B_re, B_im};

  v8f acc = {0.f, 0.f, 0.f, 0.f, 0.f, 0.f, 0.f, 0.f};

  for (int kk = 0; kk < 256; kk += 4) {
    const int k = kbase + kk + khalf;
#pragma unroll
    for (int m = 0; m < 4; ++m) {
      // A tile: 2 consecutive fp32 per lane (b64 load)
      v2f a = *(const v2f*)(mats[m] + (size_t)row * NN + k);
      // B operand from packed vector table (hot in L2; 128 KB total)
      const float* Vm = V + (size_t)m * NN * 2;
      float bx = Vm[(size_t)2 * k + bsel];
      float by = Vm[(size_t)2 * (k + 1) + bsel];
      v2f b;
      b.x = bact ? bx : 0.0f;
      b.y = bact ? by : 0.0f;
      acc = __builtin_amdgcn_wmma_f32_16x16x4_f32(
          /*neg_a=*/false, a, /*neg_b=*/false, b,
          /*c_mod=*/(short)0, acc, /*reuse_a=*/false, /*reuse_b=*/false);
    }
  }

  // D layout: lane n(0..15) VGPR v -> D[M=v,   N=n]
  //           lane n(16..31) VGPR v -> D[M=v+8, N=n-16]
  // We need columns N=0 (re) and N=1 (im).
  if (nsel < 2) {
    const int mrow0 = row0 + (lane >> 4) * 8;
#pragma unroll
    for (int v = 0; v < 8; ++v) {
      partials[(((size_t)(mrow0 + v)) * 16 + chunk) * 2 + nsel] = acc[v];
    }
  }
}

// ---------------------------------------------------------------------------
// Kernel 3: deterministic fixed-order reduction over the 16 K-chunks, *0.5.
// zi[m*2+0]=re, zi[m*2+1]=im
// ---------------------------------------------------------------------------
__global__ void reduce_zi(const float* __restrict__ partials,
                          float* __restrict__ zi) {
  int m = blockIdx.x * blockDim.x + threadIdx.x;
  if (m >= NN) return;
  float sr = 0.f, si = 0.f;
#pragma unroll
  for (int c = 0; c < 16; ++c) {
    sr += partials[((size_t)m * 16 + c) * 2 + 0];
    si += partials[((size_t)m * 16 + c) * 2 + 1];
  }
  zi[2 * m + 0] = sr * 0.5f;
  zi[2 * m + 1] = si * 0.5f;
}

// ---------------------------------------------------------------------------
// Kernel 4: streaming assembly.  Outputs are complex64 (interleaved re,im),
// concatenated in return order: y_t (4096x4096) then z_j (4096x4096).
//   z_j[m,j] = (j<4095) ? z[m,j+1] : z_i[m]
//   y_t[m,j] = z_i[m] + z_j[m,j]
// ---------------------------------------------------------------------------
__global__ __launch_bounds__(256) void assemble(
    const float* __restrict__ z_re, const float* __restrict__ z_im,
    const float* __restrict__ zi, float* __restrict__ out) {
  size_t id = (size_t)blockIdx.x * blockDim.x + threadIdx.x;
  int m = (int)(id >> 12);
  int j = (int)(id & (NN - 1));
  float zir = zi[2 * m + 0];
  float zii = zi[2 * m + 1];
  float zr, zm;
  if (j < NN - 1) {
    zr = z_re[(size_t)m * NN + j + 1];
    zm = z_im[(size_t)m * NN + j + 1];
  } else {
    zr = zir;
    zm = zii;
  }
  float2* o = (float2*)out;
  o[id]                        = make_float2(zir + zr, zii + zm);  // y_t
  o[(size_t)NN * NN + id]      = make_float2(zr, zm);              // z_j
}

// ---------------------------------------------------------------------------
// Launch. Inputs (dict order): x_re, x_im, z_re, z_im, A_re, A_im, B_re, B_im
// ws layout (floats): V[4*4096*2] @0 | partials[4096*16*2] @32768 | zi[8192] @163840
// total 172032 floats = 688 KB.
// ---------------------------------------------------------------------------
extern "C" void kernel_launch(void* const* d_in, const int* in_sizes, int n_in,
                              void* d_out, int out_size, void* d_ws,
                              size_t ws_size, hipStream_t stream) {
  const float* x_re = (const float*)d_in[0];
  const float* x_im = (const float*)d_in[1];
  const float* z_re = (const float*)d_in[2];
  const float* z_im = (const float*)d_in[3];
  const float* A_re = (const float*)d_in[4];
  const float* A_im = (const float*)d_in[5];
  const float* B_re = (const float*)d_in[6];
  const float* B_im = (const float*)d_in[7];

  float* V        = (float*)d_ws;
  float* partials = V + (size_t)4 * NN * 2;       // +32768
  float* zi       = partials + (size_t)NN * 16 * 2;  // +131072
  float* out      = (float*)d_out;

  build_vectors<<<NN / 256, 256, 0, stream>>>(x_re, x_im, z_re, z_im, V);

  matvec_wmma<<<dim3(NN / 16, 2), 256, 0, stream>>>(A_re, A_im, B_re, B_im, V,
                                                    partials);

  reduce_zi<<<NN / 256, 256, 0, stream>>>(partials, zi);

  assemble<<<(unsigned)(((size_t)NN * NN) / 256), 256, 0, stream>>>(z_re, z_im,
                                                                    zi, out);
}